// InvRadonFourierLayer_31224412242065
// MI455X (gfx1250) — compile-verified
//
#include <hip/hip_runtime.h>
#include <math.h>

// Problem constants (fixed by the reference setup)
#define H_IN  768    // FFT/filter length (rows of sinogram)
#define W_ANG 360    // number of projection angles
#define NB    16     // batch
#define DOUT  768    // output image side

typedef float v2f __attribute__((ext_vector_type(2)));
typedef float v8f __attribute__((ext_vector_type(8)));

// Async global->LDS copy of one dword per lane (CDNA5 ASYNCcnt path).
// lds_off: LDS byte offset (low 32 bits of the generic shared pointer),
// gaddr:   64-bit global byte address.
__device__ __forceinline__ void async_load_b32_to_lds(unsigned lds_off,
                                                      unsigned long long gaddr) {
    asm volatile("global_load_async_to_lds_b32 %0, %1, off"
                 :: "v"(lds_off), "v"(gaddr)
                 : "memory");
}

__device__ __forceinline__ void wait_asynccnt0() {
    asm volatile("s_wait_asynccnt 0x0" ::: "memory");
}

// ---------------------------------------------------------------------------
// Kernel 1: spatial ramp-filter kernel  g[n] = (1/H) * sum_k hG[k]*cos(2*pi*k*n/H)
// (hG is real and even => IFFT is real; frequency-domain multiply == circular
//  convolution with g). 768x768 cosines, negligible cost.
// ---------------------------------------------------------------------------
__global__ void build_filter_kernel(const float* __restrict__ hG,
                                    float* __restrict__ g) {
    int n = blockIdx.x * blockDim.x + threadIdx.x;
    if (n >= H_IN) return;
    const float w0 = 6.28318530717958647692f / (float)H_IN;
    float acc = 0.0f;
    for (int k = 0; k < H_IN; ++k) {
        int ph = (k * n) % H_IN;            // keep cos argument in [0, 2*pi)
        acc = fmaf(hG[k], cosf(w0 * (float)ph), acc);
    }
    g[n] = acc * (1.0f / (float)H_IN);
}

// ---------------------------------------------------------------------------
// Kernel 2: circulant-filter GEMM on the WMMA pipe (f32 16x16x4).
//   F[w][h][n] = sum_k g[(h-k) mod H] * X[n][k][w]
// One workgroup per angle w. B-panel (768 x 16 batch columns, 49 KB) and g
// (3 KB) staged into LDS with GLOBAL_LOAD_ASYNC_TO_LDS_B32 (ASYNCcnt), then
// 8 waves x 6 h-tiles = 48 tiles. Per K-step: one shared B fragment, 6 A
// fragments, 6 v_wmma_f32_16x16x4_f32 into independent accumulators.
// ---------------------------------------------------------------------------
__global__ __launch_bounds__(256)
void filter_gemm_kernel(const float* __restrict__ X,   // [NB][H_IN][W_ANG]
                        const float* __restrict__ g,   // [H_IN]
                        float* __restrict__ F) {       // [W_ANG][H_IN][NB]
    __shared__ float Xs[H_IN * NB];   // [k][n]
    __shared__ float gs[H_IN];

    const int w   = blockIdx.x;
    const int tid = threadIdx.x;

    // Async-stage B panel: Xs[k][n] = X[n][k][w].  12288/256 = 48 uniform
    // iterations -> EXEC all-ones, no divergence around the async ops.
    for (int idx = tid; idx < H_IN * NB; idx += 256) {
        int k = idx >> 4;
        int n = idx & 15;
        const float* src = X + (size_t)n * (H_IN * W_ANG) + (size_t)k * W_ANG + w;
        async_load_b32_to_lds((unsigned)(uintptr_t)&Xs[idx],
                              (unsigned long long)(uintptr_t)src);
    }
    // Async-stage the filter kernel (768/256 = 3 uniform iterations).
    for (int i = tid; i < H_IN; i += 256) {
        async_load_b32_to_lds((unsigned)(uintptr_t)&gs[i],
                              (unsigned long long)(uintptr_t)(g + i));
    }
    wait_asynccnt0();      // this wave's async transfers landed in LDS
    __syncthreads();       // all waves' transfers visible

    const int lane  = tid & 31;
    const int wave  = tid >> 5;
    const int l15   = lane & 15;
    const int khalf = (lane >> 4) * 2;        // 0 for lanes 0-15, 2 for 16-31

    v8f acc[6];
    #pragma unroll
    for (int j = 0; j < 6; ++j) acc[j] = (v8f){0.f,0.f,0.f,0.f,0.f,0.f,0.f,0.f};

    for (int k0 = 0; k0 < H_IN; k0 += 4) {
        const int kb = k0 + khalf;
        // B fragment (4x16): VGPR0 = rows {k0, k0+2}, VGPR1 = rows {k0+1, k0+3}
        v2f b;
        b.x = Xs[kb * NB + l15];
        b.y = Xs[(kb + 1) * NB + l15];
        #pragma unroll
        for (int j = 0; j < 6; ++j) {
            // A fragment (16x4): lane holds row M = h0+l15, cols kb, kb+1
            const int M  = (wave * 6 + j) * 16 + l15;
            int d  = M - kb + H_IN;                    // [2, 1535]
            int d0 = d  - ((d  >= H_IN) ? H_IN : 0);   // (M-kb)   mod H
            int dm = d - 1;
            int d1 = dm - ((dm >= H_IN) ? H_IN : 0);   // (M-kb-1) mod H
            v2f a;
            a.x = gs[d0];
            a.y = gs[d1];
            acc[j] = __builtin_amdgcn_wmma_f32_16x16x4_f32(
                         false, a, false, b, (short)0, acc[j], false, false);
        }
    }

    // C/D layout: VGPR v holds M = h0 + v + 8*(lane>>4), N = lane&15
    float* Fw = F + (size_t)w * (H_IN * NB);
    for (int j = 0; j < 6; ++j) {
        const int mbase = (wave * 6 + j) * 16 + (lane >> 4) * 8;
        #pragma unroll
        for (int v = 0; v < 8; ++v) {
            Fw[(size_t)(mbase + v) * NB + l15] = acc[j][v];
        }
    }
}

// ---------------------------------------------------------------------------
// Kernel 3: backprojection. One thread = one output pixel, 16 batch
// accumulators in registers. Interp index/weight shared across batch; the two
// needed rows of F[w] are 2 x 64 B contiguous (float4 loads). Entire F
// (17.7 MB) fits in the 192 MB L2, so gathers are cache hits.
// ---------------------------------------------------------------------------
__global__ __launch_bounds__(256)
void backproject_kernel(const float* __restrict__ F,   // [W_ANG][H_IN][NB]
                        float* __restrict__ out) {     // [NB][DOUT][DOUT]
    const int j = blockIdx.x * 16 + (threadIdx.x & 15);   // column index (ypr)
    const int i = blockIdx.y * 16 + (threadIdx.x >> 4);   // row index    (xpr)
    const float rx = (float)(i - DOUT / 2);
    const float ry = (float)(j - DOUT / 2);

    // py = (t/half + 1) * 0.5 * (H-1) = t*(H-1)/H + (H-1)/2   (half = H/2)
    const float scale_t = (float)(H_IN - 1) / (float)H_IN;
    const float off     = 0.5f * (float)(H_IN - 1);
    const float dth     = 3.14159265358979323846f / 360.0f;  // pi/180 * 0.5deg

    float acc[NB];
    #pragma unroll
    for (int n = 0; n < NB; ++n) acc[n] = 0.0f;

    for (int w = 0; w < W_ANG; ++w) {
        float s, c;
        __sincosf(dth * (float)w, &s, &c);
        const float t  = ry * c - rx * s;
        const float py = fmaf(t, scale_t, off);
        const float fi = floorf(py);
        const int   i0 = (int)fi;
        const float w1 = py - fi;
        const int   i1 = i0 + 1;
        const float wa = (i0 >= 0 && i0 < H_IN) ? (1.0f - w1) : 0.0f;
        const float wb = (i1 >= 0 && i1 < H_IN) ? w1 : 0.0f;
        const int i0c = min(max(i0, 0), H_IN - 1);
        const int i1c = min(max(i1, 0), H_IN - 1);

        const float4* r0 = (const float4*)(F + ((size_t)w * H_IN + i0c) * NB);
        const float4* r1 = (const float4*)(F + ((size_t)w * H_IN + i1c) * NB);
        if (w + 1 < W_ANG) {   // warm L0/L2 for next angle (global_prefetch_b8)
            __builtin_prefetch((const void*)(F + ((size_t)(w + 1) * H_IN + i0c) * NB), 0, 0);
        }
        #pragma unroll
        for (int q = 0; q < 4; ++q) {
            float4 a4 = r0[q];
            float4 b4 = r1[q];
            acc[4*q + 0] += wa * a4.x + wb * b4.x;
            acc[4*q + 1] += wa * a4.y + wb * b4.y;
            acc[4*q + 2] += wa * a4.z + wb * b4.z;
            acc[4*q + 3] += wa * a4.w + wb * b4.w;
        }
    }

    const float sc = 3.14159265358979323846f / (2.0f * (float)W_ANG);
    float* o = out + (size_t)i * DOUT + j;
    #pragma unroll
    for (int n = 0; n < NB; ++n)
        o[(size_t)n * (DOUT * DOUT)] = acc[n] * sc;
}

// ---------------------------------------------------------------------------
extern "C" void kernel_launch(void* const* d_in, const int* in_sizes, int n_in,
                              void* d_out, int out_size, void* d_ws, size_t ws_size,
                              hipStream_t stream) {
    const float* radon = (const float*)d_in[0];   // [16][1][768][360] f32
    const float* hG    = (const float*)d_in[1];   // [768] f32
    (void)in_sizes; (void)n_in; (void)out_size; (void)ws_size;

    float* g = (float*)d_ws;                      // 768 floats
    float* F = (float*)d_ws + 1024;               // [360][768][16] = 17.7 MB

    build_filter_kernel<<<(H_IN + 255) / 256, 256, 0, stream>>>(hG, g);
    filter_gemm_kernel<<<W_ANG, 256, 0, stream>>>(radon, g, F);
    dim3 grid(DOUT / 16, DOUT / 16);
    backproject_kernel<<<grid, 256, 0, stream>>>(F, (float*)d_out);
}